// IterativeModelWithFirst_Nei_2937757631151
// MI455X (gfx1250) — compile-verified
//
#include <hip/hip_runtime.h>
#include <hip/hip_bf16.h>

#define Bq   2
#define Cc   1024
#define HWd  224
#define Nb   80
#define Dt   1026
#define Dff  2048
#define Tn   320        // tokens per batch: qf|bt|fqf|nf = 4*80
#define NFK  18432      // 9 offsets * 1024 ch * 2 levels

typedef float v2f __attribute__((ext_vector_type(2)));
typedef float v8f __attribute__((ext_vector_type(8)));

#define WMMA_F32(A_, B_, C_) \
    __builtin_amdgcn_wmma_f32_16x16x4_f32(false, (A_), false, (B_), (short)0, (C_), false, false)

// ---------------------------------------------------------------------------
// Batched GEMM on V_WMMA_F32_16X16X4_F32 (true FP32 matrix pipe).
// C[b] = act( scale * A[b] @ B[b](^T) + bias )
// One wave owns a 16x64 tile (4 sub-tiles of 16x16); 8 waves / 256-thr block.
// M/N tails handled by CLAMPING row/col indices (WMMA element (m,n) depends
// only on A-row m / B-col n, and OOB elements are never stored) -> all loads
// in the main K loop are unconditional and branchless. K tail (K%4) is one
// extra WMMA with A masked to zero (B clamped; zero A annihilates B).
// ---------------------------------------------------------------------------
template<int TB>
__global__ void wmma_gemm_f32(const float* __restrict__ A, int lda, long strideA,
                              const float* __restrict__ Bm, int ldb, long strideB,
                              const float* __restrict__ bias,
                              float* __restrict__ Cm, int ldc, long strideC,
                              int M, int N, int K, float scale, int relu)
{
    const int batch = blockIdx.y;
    const float* Ab = A  + (long)batch * strideA;
    const float* Bb = Bm + (long)batch * strideB;
    float*       Cb = Cm + (long)batch * strideC;

    const int tilesN = (N + 63) >> 6;
    const int tilesM = (M + 15) >> 4;
    const int wave   = blockIdx.x * (blockDim.x >> 5) + (threadIdx.x >> 5);
    if (wave >= tilesM * tilesN) return;          // whole-wave exit
    const int tm = wave / tilesN;
    const int tn = wave % tilesN;

    const int lane = threadIdx.x & 31;
    const int half = lane >> 4;                   // K sub-group per ISA layout
    const int mn   = lane & 15;

    const int row  = tm * 16 + mn;
    const int rowC = row < M ? row : M - 1;       // clamp, never masked
    int col[4], colC[4];
#pragma unroll
    for (int t = 0; t < 4; ++t) {
        col[t]  = tn * 64 + t * 16 + mn;
        colC[t] = col[t] < N ? col[t] : N - 1;
    }

    const int Kmain = K & ~3;
    const float* ap = Ab + (long)rowC * lda + 2 * half;

    v8f acc[4] = {{}, {}, {}, {}};

    if (TB) {
        const float* bp[4];
#pragma unroll
        for (int t = 0; t < 4; ++t) bp[t] = Bb + (long)colC[t] * ldb + 2 * half;
        for (int k0 = 0; k0 < Kmain; k0 += 4) {
            v2f a = *(const v2f*)(ap + k0);               // global_load_b64
#pragma unroll
            for (int t = 0; t < 4; ++t) {
                v2f b = *(const v2f*)(bp[t] + k0);        // contiguous along K
                acc[t] = WMMA_F32(a, b, acc[t]);
            }
        }
    } else {
        for (int k0 = 0; k0 < Kmain; k0 += 4) {
            v2f a = *(const v2f*)(ap + k0);               // global_load_b64
            const float* br0 = Bb + (long)(k0 + 2 * half) * ldb;
            const float* br1 = br0 + ldb;
#pragma unroll
            for (int t = 0; t < 4; ++t) {
                v2f b; b.x = br0[colC[t]]; b.y = br1[colC[t]];
                acc[t] = WMMA_F32(a, b, acc[t]);
            }
        }
    }

    if (K & 3) {                                          // K tail: mask A only
        const int ka  = Kmain + 2 * half;
        const int kc0 = ka     < K ? ka     : K - 1;
        const int kc1 = ka + 1 < K ? ka + 1 : K - 1;
        const float a0 = Ab[(long)rowC * lda + kc0];
        const float a1 = Ab[(long)rowC * lda + kc1];
        v2f a;
        a.x = (ka     < K) ? a0 : 0.f;
        a.y = (ka + 1 < K) ? a1 : 0.f;
#pragma unroll
        for (int t = 0; t < 4; ++t) {
            v2f b;
            if (TB) {
                b.x = Bb[(long)colC[t] * ldb + kc0];
                b.y = Bb[(long)colC[t] * ldb + kc1];
            } else {
                b.x = Bb[(long)kc0 * ldb + colC[t]];
                b.y = Bb[(long)kc1 * ldb + colC[t]];
            }
            acc[t] = WMMA_F32(a, b, acc[t]);
        }
    }

#pragma unroll
    for (int t = 0; t < 4; ++t) {
#pragma unroll
        for (int r = 0; r < 8; ++r) {                     // D: VGPR r -> M = 8*half + r
            const int m = tm * 16 + half * 8 + r;
            const int n = col[t];
            if (m < M && n < N) {
                float v = acc[t][r] * scale + (bias ? bias[n] : 0.f);
                if (relu) v = v > 0.f ? v : 0.f;
                Cb[(long)m * ldc + n] = v;
            }
        }
    }
}

// ---------------------------------------------------------------------------
// _best_shift_roll: one block per batch, argmin over 80 cyclic shifts.
// ---------------------------------------------------------------------------
__global__ void best_shift_kernel(const int* __restrict__ prev, const int* __restrict__ first,
                                  int* __restrict__ fb)
{
    const int b = blockIdx.x;
    const int* pb = prev  + b * Nb * 2;
    const int* fd = first + b * Nb * 2;
    int*       ob = fb    + b * Nb * 2;
    __shared__ int sd[128], si[128];
    const int t = threadIdx.x;
    int d = 0x7FFFFFFF;
    if (t < Nb) {
        int s = 0;
        for (int j = 0; j < Nb; ++j) {
            int m = j - t; m %= Nb; if (m < 0) m += Nb;
            s += abs(pb[j*2+0] - fd[m*2+0]) + abs(pb[j*2+1] - fd[m*2+1]);
        }
        d = s;
    }
    sd[t] = d; si[t] = t;
    __syncthreads();
    for (int s = 64; s > 0; s >>= 1) {
        if (t < s && sd[t + s] < sd[t]) { sd[t] = sd[t + s]; si[t] = si[t + s]; }
        __syncthreads();
    }
    const int best = si[0];
    if (t < Nb) {
        int m = t - best; m %= Nb; if (m < 0) m += Nb;
        ob[t*2+0] = fd[m*2+0];
        ob[t*2+1] = fd[m*2+1];
    }
}

// out (B,Nb,Cc): out[b,n,c] = feats[b,c,y,x]
__global__ void gather_kernel(const float* __restrict__ feats, const int* __restrict__ bnd,
                              float* __restrict__ out)
{
    long idx = (long)blockIdx.x * blockDim.x + threadIdx.x;
    if (idx >= (long)Bq * Nb * Cc) return;
    int c = idx % Cc; long t = idx / Cc;
    int n = t % Nb;   int b = t / Nb;
    int y = bnd[(b*Nb+n)*2+0], x = bnd[(b*Nb+n)*2+1];
    out[idx] = feats[(((long)b * Cc + c) * HWd + y) * HWd + x];
}

// Neighbor features (B,Nb,18432), lvl-2 4x4 pooling computed on the fly.
__global__ void neighbor_kernel(const float* __restrict__ feats, const int* __restrict__ bl,
                                float* __restrict__ out)
{
    long idx = (long)blockIdx.x * blockDim.x + threadIdx.x;
    if (idx >= (long)Bq * Nb * NFK) return;
    int j = idx % NFK; long t = idx / NFK;
    int n = t % Nb;    int b = t / Nb;
    const int y = bl[(b*Nb+n)*2+0], x = bl[(b*Nb+n)*2+1];
    const int lvl = j / (9 * Cc);
    const int jj  = j % (9 * Cc);
    const int off = jj / Cc;
    const int c   = jj % Cc;
    const int d0 = off / 3 - 1, d1 = off % 3 - 1;
    const float* f = feats + ((long)b * Cc + c) * (HWd * HWd);
    float v = 0.f;
    if (lvl == 0) {
        const int py = y + d0, px = x + d1;
        if (py >= 0 && py < HWd && px >= 0 && px < HWd) v = f[py * HWd + px];
    } else {
        const int qy = (y >> 2) + d0, qx = (x >> 2) + d1;
        if (qy >= 0 && qy < 56 && qx >= 0 && qx < 56) {
            float s = 0.f;
            for (int a = 0; a < 4; ++a)
                for (int e = 0; e < 4; ++e)
                    s += f[(qy*4 + a) * HWd + qx*4 + e];
            v = s * 0.0625f;
        }
    }
    out[idx] = v;
}

// tok (B,320,1026) = concat rows [qf | (bf,cb) | fqf | nf]
__global__ void build_tok_kernel(const float* __restrict__ qf, const float* __restrict__ bf,
                                 const float* __restrict__ cb, const float* __restrict__ fqf,
                                 const float* __restrict__ nf, float* __restrict__ tok)
{
    long idx = (long)blockIdx.x * blockDim.x + threadIdx.x;
    if (idx >= (long)Bq * Tn * Dt) return;
    int j = idx % Dt; long t = idx / Dt;
    int r = t % Tn;   int b = t / Tn;
    int seg = r / Nb, n = r % Nb;
    float v;
    if      (seg == 0) v = qf [((long)b*Nb + n)*Dt + j];
    else if (seg == 1) v = (j < Cc) ? bf[((long)b*Nb + n)*Cc + j] : cb[(b*Nb + n)*2 + (j - Cc)];
    else if (seg == 2) v = fqf[((long)b*Nb + n)*Dt + j];
    else               v = nf [((long)b*Nb + n)*Dt + j];
    tok[idx] = v;
}

// Row LayerNorm, optional residual input, optional fused sinusoidal PE add.
__global__ void ln_kernel(const float* __restrict__ X, const float* __restrict__ R,
                          const float* __restrict__ g, const float* __restrict__ be,
                          float* __restrict__ Y, int D, int rowsPerBatch, int addPE)
{
    const long rowG = blockIdx.x;
    const float* x = X + rowG * D;
    const float* r = R ? R + rowG * D : nullptr;
    float*       y = Y + rowG * D;
    const int p = (int)(rowG % rowsPerBatch);
    __shared__ float s1[256], s2[256];
    float sum = 0.f, sq = 0.f;
    for (int j = threadIdx.x; j < D; j += blockDim.x) {
        float v = x[j] + (r ? r[j] : 0.f);
        sum += v; sq += v * v;
    }
    s1[threadIdx.x] = sum; s2[threadIdx.x] = sq;
    __syncthreads();
    for (int s = blockDim.x / 2; s > 0; s >>= 1) {
        if (threadIdx.x < s) { s1[threadIdx.x] += s1[threadIdx.x+s]; s2[threadIdx.x] += s2[threadIdx.x+s]; }
        __syncthreads();
    }
    const float mean = s1[0] / D;
    const float var  = s2[0] / D - mean * mean;
    const float inv  = rsqrtf(var + 1e-5f);
    for (int j = threadIdx.x; j < D; j += blockDim.x) {
        float v = x[j] + (r ? r[j] : 0.f);
        float o = (v - mean) * inv * g[j] + be[j];
        if (addPE) {
            const float freq = __expf(-(float)(j & ~1) * (9.210340371976184f / (float)D));
            const float ang  = (float)p * freq;
            o += (j & 1) ? __cosf(ang) : __sinf(ang);
        }
        y[j] = o;
    }
}

__global__ void softmax_kernel(float* __restrict__ S, int N)
{
    float* row = S + (long)blockIdx.x * N;
    __shared__ float red[256];
    const int t = threadIdx.x;
    float mx = -1e30f;
    for (int j = t; j < N; j += blockDim.x) mx = fmaxf(mx, row[j]);
    red[t] = mx; __syncthreads();
    for (int s = blockDim.x/2; s > 0; s >>= 1) { if (t < s) red[t] = fmaxf(red[t], red[t+s]); __syncthreads(); }
    mx = red[0]; __syncthreads();
    float sum = 0.f;
    for (int j = t; j < N; j += blockDim.x) { float e = __expf(row[j] - mx); row[j] = e; sum += e; }
    red[t] = sum; __syncthreads();
    for (int s = blockDim.x/2; s > 0; s >>= 1) { if (t < s) red[t] += red[t+s]; __syncthreads(); }
    const float invs = 1.f / red[0];
    for (int j = t; j < N; j += blockDim.x) row[j] *= invs;
}

// b_off = tok[:, 80:160] @ Wb(1026x2) + bb    (tiny: one thread per output)
__global__ void boff_kernel(const float* __restrict__ tok, const float* __restrict__ Wb,
                            const float* __restrict__ bb, float* __restrict__ boff)
{
    int idx = blockIdx.x * blockDim.x + threadIdx.x;
    if (idx >= Bq * Nb * 2) return;
    const int k = idx & 1, n = (idx >> 1) % Nb, b = idx / (Nb * 2);
    const float* row = tok + ((long)b * Tn + Nb + n) * Dt;
    float s = bb[k];
    for (int c = 0; c < Dt; ++c) s += row[c] * Wb[c * 2 + k];
    boff[idx] = s;
}

__global__ void cb_init_kernel(const int* __restrict__ prev, float* __restrict__ cb)
{
    int i = blockIdx.x * blockDim.x + threadIdx.x;
    if (i < Bq * Nb * 2) cb[i] = (float)prev[i];
}

__global__ void bl_kernel(const float* __restrict__ cb, int* __restrict__ bl)
{
    int i = blockIdx.x * blockDim.x + threadIdx.x;
    if (i < Bq * Nb * 2) bl[i] = (int)cb[i];
}

__global__ void update_cb_kernel(float* __restrict__ cb, const float* __restrict__ boff,
                                 float* __restrict__ outp)
{
    int i = blockIdx.x * blockDim.x + threadIdx.x;
    if (i >= Bq * Nb * 2) return;
    float v = cb[i] + boff[i];
    v = fminf(fmaxf(v, 0.f), 223.f);
    cb[i] = v;
    outp[i] = v;
}

__global__ void add_kernel(float* __restrict__ a, const float* __restrict__ b, long n)
{
    long i = (long)blockIdx.x * blockDim.x + threadIdx.x;
    if (i < n) a[i] += b[i];
}

// ---------------------------------------------------------------------------
extern "C" void kernel_launch(void* const* d_in, const int* in_sizes, int n_in,
                              void* d_out, int out_size, void* d_ws, size_t ws_size,
                              hipStream_t stream)
{
    const float* firstF = (const float*)d_in[0];
    const float* preF   = (const float*)d_in[1];
    const float* currF  = (const float*)d_in[2];
    const int*   prevB  = (const int*)  d_in[3];
    const int*   firstB = (const int*)  d_in[4];
    const float* Wq   = (const float*)d_in[5];  const float* bq    = (const float*)d_in[6];
    const float* gq   = (const float*)d_in[7];  const float* bq_ln = (const float*)d_in[8];
    const float* Wn   = (const float*)d_in[9];  const float* bn    = (const float*)d_in[10];
    const float* gn   = (const float*)d_in[11]; const float* bn_ln = (const float*)d_in[12];
    const float* g_ln = (const float*)d_in[13]; const float* b_ln  = (const float*)d_in[14];
    const float* Wqkv = (const float*)d_in[15]; const float* bqkv  = (const float*)d_in[16];
    const float* Wo   = (const float*)d_in[17]; const float* bo    = (const float*)d_in[18];
    const float* g1   = (const float*)d_in[19]; const float* b1n   = (const float*)d_in[20];
    const float* W1   = (const float*)d_in[21]; const float* b1f   = (const float*)d_in[22];
    const float* W2   = (const float*)d_in[23]; const float* b2f   = (const float*)d_in[24];
    const float* g2   = (const float*)d_in[25]; const float* b2n   = (const float*)d_in[26];
    const float* Wqo  = (const float*)d_in[27]; const float* bqo   = (const float*)d_in[28];
    const float* Wb   = (const float*)d_in[29]; const float* bb    = (const float*)d_in[30];

    // ---- workspace bump allocator (floats; all sizes even -> 8B alignment) ----
    float* w = (float*)d_ws;
    size_t off = 0;
    auto alloc = [&](size_t n) { float* p = w + off; off += n; return p; };
    int*   fbI   = (int*)alloc(Bq*Nb*2);
    int*   blI   = (int*)alloc(Bq*Nb*2);
    float* cb    = alloc(Bq*Nb*2);
    float* boff  = alloc(Bq*Nb*2);
    float* gtmp  = alloc((size_t)Bq*Nb*Cc);
    float* bfB   = alloc((size_t)Bq*Nb*Cc);
    float* lin   = alloc((size_t)Bq*Nb*Dt);
    float* qf    = alloc((size_t)Bq*Nb*Dt);
    float* fqf   = alloc((size_t)Bq*Nb*Dt);
    float* qoff  = alloc((size_t)Bq*Nb*Dt);
    float* nfOut = alloc((size_t)Bq*Nb*Dt);
    float* nfIn  = alloc((size_t)Bq*Nb*NFK);
    float* tokA  = alloc((size_t)Bq*Tn*Dt);   // raw concat, later reused as tok4
    float* tok2  = alloc((size_t)Bq*Tn*Dt);   // post LN+PE
    float* attn  = alloc((size_t)Bq*Tn*Dt);
    float* tok3  = alloc((size_t)Bq*Tn*Dt);
    float* ffo   = alloc((size_t)Bq*Tn*Dt);
    float* av    = alloc((size_t)Bq*Tn*Dt);
    float* qkv   = alloc((size_t)Bq*Tn*3*Dt);
    float* ffh   = alloc((size_t)Bq*Tn*Dff);
    float* scr   = alloc((size_t)Bq*Tn*Tn);

    auto gemm = [&](const float* A, int lda, long sA,
                    const float* Bm, int ldb, long sB, int tB, const float* bias,
                    float* Cm, int ldc, long sC, int M, int N, int K,
                    float scale, int relu) {
        int tiles = ((M + 15) / 16) * ((N + 63) / 64);
        dim3 grid((tiles + 7) / 8, Bq);
        if (tB)
            wmma_gemm_f32<1><<<grid, 256, 0, stream>>>(A, lda, sA, Bm, ldb, sB, bias,
                                                       Cm, ldc, sC, M, N, K, scale, relu);
        else
            wmma_gemm_f32<0><<<grid, 256, 0, stream>>>(A, lda, sA, Bm, ldb, sB, bias,
                                                       Cm, ldc, sC, M, N, K, scale, relu);
    };
    const float attScale = 1.f / sqrtf((float)Dt);
    const int small = Bq * Nb * 2;

    // fb = best cyclic shift of first_boundary; cb = float(previous_boundary)
    best_shift_kernel<<<Bq, 128, 0, stream>>>(prevB, firstB, fbI);
    cb_init_kernel<<<(small + 127)/128, 128, 0, stream>>>(prevB, cb);

    // qf = LN(gather(pre, prev) @ Wq + bq); fqf = LN(gather(first, fb) @ Wq + bq)
    {
        long tot = (long)Bq*Nb*Cc;
        gather_kernel<<<(tot + 255)/256, 256, 0, stream>>>(preF, prevB, gtmp);
        gemm(gtmp, Cc, (long)Nb*Cc, Wq, Dt, 0, 0, bq, lin, Dt, (long)Nb*Dt, Nb, Dt, Cc, 1.f, 0);
        ln_kernel<<<Bq*Nb, 256, 0, stream>>>(lin, nullptr, gq, bq_ln, qf, Dt, Nb, 0);
        gather_kernel<<<(tot + 255)/256, 256, 0, stream>>>(firstF, fbI, gtmp);
        gemm(gtmp, Cc, (long)Nb*Cc, Wq, Dt, 0, 0, bq, lin, Dt, (long)Nb*Dt, Nb, Dt, Cc, 1.f, 0);
        ln_kernel<<<Bq*Nb, 256, 0, stream>>>(lin, nullptr, gq, bq_ln, fqf, Dt, Nb, 0);
    }

    for (int it = 0; it < 3; ++it) {
        bl_kernel<<<(small + 127)/128, 128, 0, stream>>>(cb, blI);

        // neighbor features -> nf = LN(nfIn @ Wn + bn)
        long totN = (long)Bq*Nb*NFK;
        neighbor_kernel<<<(totN + 255)/256, 256, 0, stream>>>(currF, blI, nfIn);
        gemm(nfIn, NFK, (long)Nb*NFK, Wn, Dt, 0, 0, bn, lin, Dt, (long)Nb*Dt, Nb, Dt, NFK, 1.f, 0);
        ln_kernel<<<Bq*Nb, 256, 0, stream>>>(lin, nullptr, gn, bn_ln, nfOut, Dt, Nb, 0);

        // bf gather + token assembly + LN + PE
        long totG = (long)Bq*Nb*Cc;
        gather_kernel<<<(totG + 255)/256, 256, 0, stream>>>(currF, blI, bfB);
        long totT = (long)Bq*Tn*Dt;
        build_tok_kernel<<<(totT + 255)/256, 256, 0, stream>>>(qf, bfB, cb, fqf, nfOut, tokA);
        ln_kernel<<<Bq*Tn, 256, 0, stream>>>(tokA, nullptr, g_ln, b_ln, tok2, Dt, Tn, 1);

        // attention
        gemm(tok2, Dt, (long)Tn*Dt, Wqkv, 3*Dt, 0, 0, bqkv,
             qkv, 3*Dt, (long)Tn*3*Dt, Tn, 3*Dt, Dt, 1.f, 0);
        gemm(qkv, 3*Dt, (long)Tn*3*Dt, qkv + Dt, 3*Dt, (long)Tn*3*Dt, 1, nullptr,
             scr, Tn, (long)Tn*Tn, Tn, Tn, Dt, attScale, 0);
        softmax_kernel<<<Bq*Tn, 256, 0, stream>>>(scr, Tn);
        gemm(scr, Tn, (long)Tn*Tn, qkv + 2*Dt, 3*Dt, (long)Tn*3*Dt, 0, nullptr,
             av, Dt, (long)Tn*Dt, Tn, Dt, Tn, 1.f, 0);
        gemm(av, Dt, (long)Tn*Dt, Wo, Dt, 0, 0, bo, attn, Dt, (long)Tn*Dt, Tn, Dt, Dt, 1.f, 0);
        ln_kernel<<<Bq*Tn, 256, 0, stream>>>(tok2, attn, g1, b1n, tok3, Dt, Tn, 0);

        // feed-forward
        gemm(tok3, Dt, (long)Tn*Dt, W1, Dff, 0, 0, b1f, ffh, Dff, (long)Tn*Dff, Tn, Dff, Dt, 1.f, 1);
        gemm(ffh, Dff, (long)Tn*Dff, W2, Dt, 0, 0, b2f, ffo, Dt, (long)Tn*Dt, Tn, Dt, Dff, 1.f, 0);
        ln_kernel<<<Bq*Tn, 256, 0, stream>>>(tok3, ffo, g2, b2n, tokA, Dt, Tn, 0); // tok4 in tokA

        // heads + state update
        gemm(tokA, Dt, (long)Tn*Dt, Wqo, Dt, 0, 0, bqo, qoff, Dt, (long)Nb*Dt, Nb, Dt, Dt, 1.f, 0);
        boff_kernel<<<(small + 127)/128, 128, 0, stream>>>(tokA, Wb, bb, boff);
        long totQ = (long)Bq*Nb*Dt;
        add_kernel<<<(totQ + 255)/256, 256, 0, stream>>>(qf, qoff, totQ);
        update_cb_kernel<<<(small + 127)/128, 128, 0, stream>>>(cb, boff,
            (float*)d_out + (long)it * Bq * Nb * 2);
    }
    (void)in_sizes; (void)n_in; (void)out_size; (void)ws_size;
}